// GNNLayer_1778116460648
// MI455X (gfx1250) — compile-verified
//
#include <hip/hip_runtime.h>
#include <math.h>

#define NN  1024
#define DIM 128
#define LDS_STRIDE 132   // 128 + 4 pad: rows land 4 banks apart -> conflict-free

typedef __attribute__((ext_vector_type(2))) float v2f;
typedef __attribute__((ext_vector_type(4))) float v4f;   // native vector (not HIP class)
typedef __attribute__((ext_vector_type(8))) float v8f;
typedef int b128_t __attribute__((vector_size(16)));     // matches builtin's V4i

// CDNA5 async global->LDS path (ASYNCcnt-tracked, no VGPR staging), if the
// toolchain exposes the builtins; otherwise fall back to sync load+ds_store.
#if __has_builtin(__builtin_amdgcn_global_load_async_to_lds_b128) && \
    __has_builtin(__builtin_amdgcn_s_wait_asynccnt)
#define USE_ASYNC_LDS 1
#define ASYNC_B128(gsrc, ldst)                                            \
    __builtin_amdgcn_global_load_async_to_lds_b128(                       \
        (__attribute__((address_space(1))) b128_t*)(gsrc),                \
        (__attribute__((address_space(3))) b128_t*)(ldst), 0, 0)
#else
#define USE_ASYNC_LDS 0
#endif

// ---------------------------------------------------------------------------
// Kernel A: np_b[j][o] = node_features[j,:] . W_msg[0:128, o] + b_msg[o]
// ---------------------------------------------------------------------------
__global__ __launch_bounds__(128)
void gnn_nodepart_kernel(const float* __restrict__ node,
                         const float* __restrict__ Wmsg,   // [256][128]
                         const float* __restrict__ bmsg,   // [128]
                         float* __restrict__ npb)          // [N][128]
{
    const int j = blockIdx.x;
    const int o = threadIdx.x;
    __shared__ float s_node[DIM];
    s_node[o] = node[(size_t)j * DIM + o];
    __syncthreads();
    float acc = bmsg[o];
#pragma unroll 8
    for (int k = 0; k < DIM; ++k)
        acc += s_node[k] * Wmsg[(size_t)k * DIM + o];
    npb[(size_t)j * DIM + o] = acc;
}

// ---------------------------------------------------------------------------
// Kernel B (dominant): for each i:
//   aggregated[i,o] = sum_j mask[i,j] * relu( edge[i,j,:] . Wm_edge[:,o] + np_b[j,o] )
// One workgroup per i; 8 waves; wave w owns o-columns [16w, 16w+16).
// j processed in double-buffered tiles of 16 rows; GEMM via V_WMMA_F32_16X16X4_F32.
// ---------------------------------------------------------------------------
__global__ __launch_bounds__(256)
void gnn_msg_agg_kernel(const float* __restrict__ edge,   // [N][N][128]
                        const float* __restrict__ Wmsg,   // [256][128]; rows 128..255 = Wm_edge
                        const float* __restrict__ npb,    // [N][128] (node_part + b_msg)
                        const int*   __restrict__ adj,    // [N][N]
                        float* __restrict__ agg)          // [N][128]
{
    const int i      = blockIdx.x;
    const int tid    = threadIdx.x;       // 0..255
    const int wave   = tid >> 5;          // 0..7
    const int lane   = tid & 31;
    const int lane16 = lane & 15;
    const int lhalf  = lane >> 4;         // 0: K pair {0,1}, 1: {2,3}
    const int obase  = wave * 16;

    __shared__ float s_mask[NN];                                 // 4 KB
    __shared__ __align__(16) float s_edge[2][16][LDS_STRIDE];    // 16.5 KB (double buf)
    __shared__ __align__(16) float s_np[2][16][LDS_STRIDE];      // 16.5 KB

    // adjacency row -> float mask (once per workgroup)
    for (int j = tid; j < NN; j += 256)
        s_mask[j] = (adj[(size_t)i * NN + j] > 0) ? 1.0f : 0.0f;

    // Preload B fragments for Wm_edge[:, obase:obase+16) (held in VGPRs for whole kernel).
    // B (4x16 KxN): VGPR0 = K=0 (lanes 0-15) / K=2 (16-31), VGPR1 = K=1 / K=3; N = lane16.
    v2f bfrag[32];
#pragma unroll
    for (int kk = 0; kk < 32; ++kk) {
        const int krow = 128 + kk * 4 + 2 * lhalf;
        bfrag[kk].x = Wmsg[(size_t)krow * DIM + obase + lane16];
        bfrag[kk].y = Wmsg[(size_t)(krow + 1) * DIM + obase + lane16];
    }

    // Per-thread staging coordinates: thread moves 2x16B of edge + 2x16B of np_b.
    const int srow = tid >> 4;          // 0..15
    const int scol = (tid & 15) * 8;    // 0,8,...,120

    auto stage = [&](int jbase, int buf) {
        const float* esrc = edge + (((size_t)i * NN + (jbase + srow)) * DIM + scol);
        const float* nsrc = npb + ((size_t)(jbase + srow) * DIM + scol);
#if USE_ASYNC_LDS
        ASYNC_B128(esrc,     &s_edge[buf][srow][scol]);
        ASYNC_B128(esrc + 4, &s_edge[buf][srow][scol + 4]);
        ASYNC_B128(nsrc,     &s_np[buf][srow][scol]);
        ASYNC_B128(nsrc + 4, &s_np[buf][srow][scol + 4]);
#else
        // edge is a 512MB single-use stream: non-temporal to spare L2
        v4f e0 = __builtin_nontemporal_load(reinterpret_cast<const v4f*>(esrc));
        v4f e1 = __builtin_nontemporal_load(reinterpret_cast<const v4f*>(esrc + 4));
        v4f n0 = *reinterpret_cast<const v4f*>(nsrc);
        v4f n1 = *reinterpret_cast<const v4f*>(nsrc + 4);
        *reinterpret_cast<v4f*>(&s_edge[buf][srow][scol])     = e0;
        *reinterpret_cast<v4f*>(&s_edge[buf][srow][scol + 4]) = e1;
        *reinterpret_cast<v4f*>(&s_np[buf][srow][scol])       = n0;
        *reinterpret_cast<v4f*>(&s_np[buf][srow][scol + 4])   = n1;
#endif
    };

    float accsum = 0.0f;   // partial of aggregated[i, obase+lane16]

    // Prologue: fill buffer 0
    stage(0, 0);
#if USE_ASYNC_LDS
    __builtin_amdgcn_s_wait_asynccnt(0);
#endif
    __syncthreads();

    for (int t = 0; t < NN / 16; ++t) {
        const int cur = t & 1;
        // Start streaming next tile into the other buffer (overlaps WMMA chain)
        if (t + 1 < NN / 16)
            stage((t + 1) * 16, cur ^ 1);

        // Batch all A fragments into registers so DS loads pipeline ahead of WMMAs.
        // A (16x4 MxK): lane16 = M row; VGPR0 = K=0/2, VGPR1 = K=1/3.
        v2f afrag[32];
#pragma unroll
        for (int kk = 0; kk < 32; ++kk)
            afrag[kk] = *reinterpret_cast<const v2f*>(
                &s_edge[cur][lane16][kk * 4 + 2 * lhalf]);

        v8f c = {0.f, 0.f, 0.f, 0.f, 0.f, 0.f, 0.f, 0.f};
#pragma unroll
        for (int kk = 0; kk < 32; ++kk)
            c = __builtin_amdgcn_wmma_f32_16x16x4_f32(
                    false, afrag[kk], false, bfrag[kk],
                    (short)0, c, false, false);

        // Fused epilogue: relu(C + np_b) * mask, accumulate over tile rows.
        // C layout: VGPR r of this lane -> M = r + 8*lhalf, N = lane16.
#pragma unroll
        for (int r = 0; r < 8; ++r) {
            const int jj = r + 8 * lhalf;
            float m = c[r] + s_np[cur][jj][obase + lane16];
            m = (m > 0.0f) ? m : 0.0f;
            accsum += s_mask[t * 16 + jj] * m;
        }

#if USE_ASYNC_LDS
        __builtin_amdgcn_s_wait_asynccnt(0);   // our issued loads for tile t+1
#endif
        __syncthreads();                       // everyone done reading cur / staging nxt
    }

    // lanes L and L+16 hold the same o column: combine halves and store
    accsum += __shfl_xor(accsum, 16, 32);
    if (lhalf == 0)
        agg[(size_t)i * DIM + obase + lane16] = accsum;
}

// ---------------------------------------------------------------------------
// Kernel C: concat = [node[i], agg[i]];
//   updated = relu(concat @ W_upd + b_upd); gate = sigmoid(concat @ W_gate + b_gate)
//   out = gate*updated + (1-gate)*node[i,:128]
// ---------------------------------------------------------------------------
__global__ __launch_bounds__(128)
void gnn_update_kernel(const float* __restrict__ node,
                       const float* __restrict__ agg,
                       const float* __restrict__ Wupd,  const float* __restrict__ bupd,
                       const float* __restrict__ Wgate, const float* __restrict__ bgate,
                       float* __restrict__ out)
{
    const int i = blockIdx.x;
    const int o = threadIdx.x;
    __shared__ float s_c[2 * DIM];
    s_c[o]       = node[(size_t)i * DIM + o];
    s_c[DIM + o] = agg[(size_t)i * DIM + o];
    __syncthreads();
    float u = bupd[o];
    float g = bgate[o];
#pragma unroll 4
    for (int k = 0; k < 2 * DIM; ++k) {
        const float cv = s_c[k];
        u += cv * Wupd[(size_t)k * DIM + o];
        g += cv * Wgate[(size_t)k * DIM + o];
    }
    u = fmaxf(u, 0.0f);
    g = 1.0f / (1.0f + __expf(-g));
    out[(size_t)i * DIM + o] = g * u + (1.0f - g) * node[(size_t)i * DIM + o];
}

// ---------------------------------------------------------------------------
extern "C" void kernel_launch(void* const* d_in, const int* in_sizes, int n_in,
                              void* d_out, int out_size, void* d_ws, size_t ws_size,
                              hipStream_t stream)
{
    (void)in_sizes; (void)n_in; (void)out_size; (void)ws_size;
    const float* node  = (const float*)d_in[0];   // [1024][128]
    const float* edge  = (const float*)d_in[1];   // [1024][1024][128]
    const int*   adj   = (const int*)  d_in[2];   // [1024][1024]
    const float* Wmsg  = (const float*)d_in[3];   // [256][128]
    const float* bmsg  = (const float*)d_in[4];   // [128]
    const float* Wupd  = (const float*)d_in[5];   // [256][128]
    const float* bupd  = (const float*)d_in[6];   // [128]
    const float* Wgate = (const float*)d_in[7];   // [256][128]
    const float* bgate = (const float*)d_in[8];   // [128]
    float* out = (float*)d_out;                   // [1024][128]

    float* npb = (float*)d_ws;                    // [1024][128] = 512 KB
    float* agg = npb + (size_t)NN * DIM;          // [1024][128] = 512 KB

    gnn_nodepart_kernel<<<NN, 128, 0, stream>>>(node, Wmsg, bmsg, npb);
    gnn_msg_agg_kernel <<<NN, 256, 0, stream>>>(edge, Wmsg, npb, adj, agg);
    gnn_update_kernel  <<<NN, 128, 0, stream>>>(node, agg, Wupd, bupd, Wgate, bgate, out);
}